// UnfoldNd_79156247265955
// MI455X (gfx1250) — compile-verified
//
#include <hip/hip_runtime.h>
#include <cstdint>

// ---------------------------------------------------------------------------
// Unfold (im2col) for x:[16,64,128,128] f32, K=3, pad=1, stride=1, dil=1.
// out[bc*9 + k][oh*128+ow] = x[bc][oh+ki-1][ow+kj-1] (zero-padded), k=ki*3+kj.
// Pure data movement: 67 MB read + 604 MB written -> store-BW bound (~29us
// at 23.3 TB/s). Strategy: stage each 64KB plane into LDS once via CDNA5
// async global->LDS b128 copies, then stream the 9 shifted planes out with
// coalesced non-temporal b128 stores.
// ---------------------------------------------------------------------------

typedef float v4f __attribute__((ext_vector_type(4)));
typedef int   v4i __attribute__((vector_size(16)));   // matches builtin param

#define GAS __attribute__((address_space(1)))
#define LAS __attribute__((address_space(3)))

#ifndef __has_builtin
#define __has_builtin(x) 0
#endif

#if __has_builtin(__builtin_amdgcn_global_load_async_to_lds_b128)
#define ASYNC_COPY_AVAILABLE 1
#else
#define ASYNC_COPY_AVAILABLE 0
#endif

__device__ __forceinline__ void async_wait_all() {
#if __has_builtin(__builtin_amdgcn_s_wait_asynccnt)
  __builtin_amdgcn_s_wait_asynccnt(0);
#else
  asm volatile("s_wait_asynccnt 0x0" ::: "memory");
#endif
}

constexpr int Hc = 128;
constexpr int Wc = 128;
constexpr int PLANE = Hc * Wc;   // 16384 floats = 64 KB
constexpr int NT = 256;          // 8 wave32s per block
constexpr int F4_PER_PLANE = PLANE / 4;       // 4096
constexpr int ITERS = F4_PER_PLANE / NT;      // 16

__global__ __launch_bounds__(NT) void unfold_k3(const float* __restrict__ x,
                                                float* __restrict__ out) {
  __shared__ float tile[PLANE];  // one (b,c) plane, 64 KB
  const int bc = blockIdx.x;
  const float* __restrict__ src = x + (size_t)bc * PLANE;
  float* __restrict__ dst = out + (size_t)bc * (size_t)(9 * PLANE);
  const int t = threadIdx.x;

  // ---- Stage the 128x128 plane into LDS --------------------------------
#if ASYNC_COPY_AVAILABLE
  #pragma unroll
  for (int i = 0; i < ITERS; ++i) {
    const int f = i * NT + t;  // float4 index, 0..4095 (16B aligned both sides)
    __builtin_amdgcn_global_load_async_to_lds_b128(
        (GAS v4i*)(uintptr_t)(src + 4 * f),
        (LAS v4i*)(uintptr_t)(tile + 4 * f),
        /*offset=*/0, /*cpol=*/0);
  }
  async_wait_all();            // wave's own async copies complete
#else
  #pragma unroll
  for (int i = 0; i < ITERS; ++i) {
    const int f = i * NT + t;
    *(v4f*)(tile + 4 * f) = *(const v4f*)(src + 4 * f);
  }
#endif
  __syncthreads();             // all waves' copies visible

  // ---- Emit 9 shifted planes, coalesced NT b128 stores -----------------
  #pragma unroll
  for (int k = 0; k < 9; ++k) {
    const int ki = k / 3;
    const int kj = k % 3;
    float* __restrict__ plane = dst + (size_t)k * PLANE;
    #pragma unroll 4
    for (int i = 0; i < ITERS; ++i) {
      const int f   = i * NT + t;            // float4 index within plane
      const int oh  = f >> 5;                // row (32 float4s per row)
      const int c0  = ((f & 31) << 2) + kj - 1;
      const int r   = oh + ki - 1;
      const bool rin   = (unsigned)r < (unsigned)Hc;
      const int  rbase = (rin ? r : 0) << 7; // clamped -> always-valid LDS addr
      v4f v;
      #pragma unroll
      for (int e = 0; e < 4; ++e) {
        const int  c   = c0 + e;
        const bool cin = (unsigned)c < (unsigned)Wc;
        const float val = tile[rbase + (cin ? c : 0)];  // branchless ds_load
        v[e] = (rin & cin) ? val : 0.0f;                 // v_cndmask
      }
      __builtin_nontemporal_store(v, (v4f*)(plane + 4 * f));
    }
  }
}

// ---------------------------------------------------------------------------

extern "C" void kernel_launch(void* const* d_in, const int* in_sizes, int n_in,
                              void* d_out, int out_size, void* d_ws, size_t ws_size,
                              hipStream_t stream) {
  (void)in_sizes; (void)n_in; (void)out_size; (void)d_ws; (void)ws_size;
  const float* x = (const float*)d_in[0];
  float* out = (float*)d_out;
  const int BC = 16 * 64;  // B * C planes, one block each
  hipLaunchKernelGGL(unfold_k3, dim3(BC), dim3(NT), 0, stream, x, out);
}